// MipmappedTextureHighPerf_88613765251480
// MI455X (gfx1250) — compile-verified
//
#include <hip/hip_runtime.h>
#include <hip/hip_bf16.h>

// Mipmapped trilinear texture sampling, 16 channels, 8 levels (1024^2 .. 8^2).
//   Pass 1: CHW -> HWC transpose of the mip pyramid into d_ws.
//           Inbound uses the gfx1250 async global->LDS data mover at b128
//           width (VGPR-free); outbound does the 4B-granular transposed reads
//           on the LDS side (conflict-free) and stores at full b128 width.
//   Pass 2: per-sample gather: 8 taps x 4 float4 loads (one 64B cache line
//           per tap), weights folded (levelweight * bilinear), 64B store.
// The whole HWC pyramid is ~85 MB -> resident in the 192 MB L2.

#define NUM_LEVELS 8
#define CHANNELS 16

#if defined(__gfx1250__) && \
    __has_builtin(__builtin_amdgcn_global_load_async_to_lds_b32) && \
    __has_builtin(__builtin_amdgcn_s_wait_asynccnt)
#define USE_ASYNC_LDS 1
#else
#define USE_ASYNC_LDS 0
#endif

#if USE_ASYNC_LDS && __has_builtin(__builtin_amdgcn_global_load_async_to_lds_b128)
#define USE_ASYNC_B128 1
#else
#define USE_ASYNC_B128 0
#endif

#if USE_ASYNC_LDS
typedef __attribute__((address_space(1))) int async_gint;   // global b32
typedef __attribute__((address_space(3))) int async_lint;   // LDS b32
#endif
#if USE_ASYNC_B128
typedef int v4i_async __attribute__((ext_vector_type(4)));
typedef __attribute__((address_space(1))) v4i_async async_gv4i;  // global b128
typedef __attribute__((address_space(3))) v4i_async async_lv4i;  // LDS b128
#endif

struct MipPtrs { const float* m[NUM_LEVELS]; };

// Element (float) base offset of each level inside the HWC workspace.
__device__ __host__ __forceinline__ size_t level_base(int l) {
    const int bases[NUM_LEVELS] = {0, 16777216, 20971520, 22020096,
                                   22282240, 22347776, 22364160, 22368256};
    return (size_t)bases[l];
}
#define WS_TOTAL_FLOATS 22369280ull   // sum over levels of H*W*16
#define TOTAL_TILES 21845             // sum over levels of (H*W)/64

// ---------------------------------------------------------------------------
// Pass 1: CHW -> HWC transpose. One block = 16 channels x 64 pixels tile.
// LDS tile[16][68]: pitch 68 floats = 272 B keeps every 16B row chunk aligned
// and makes the transposed b32 reads bank-conflict free.
// ---------------------------------------------------------------------------
__global__ __launch_bounds__(256) void mip_transpose_kernel(MipPtrs mips,
                                                            float* __restrict__ ws) {
    __shared__ float tile[CHANNELS][68];

    // Map blockIdx.x -> (level, tile-in-level)
    int lvl = 0;
    int tb  = blockIdx.x;
    for (; lvl < NUM_LEVELS - 1; ++lvl) {
        int W  = 1024 >> lvl;
        int nt = (W * W) >> 6;          // tiles of 64 pixels
        if (tb < nt) break;
        tb -= nt;
    }
    const int W    = 1024 >> lvl;
    const int npix = W * W;
    const int pix0 = tb * 64;
    const float* __restrict__ in = mips.m[lvl];
    float* __restrict__ outp = ws + level_base(lvl);

    const int t = threadIdx.x;

    // ---- inbound: 16B per lane, global channel row -> LDS row ----
    {
        const int c  = t >> 4;          // channel 0..15
        const int xq = (t & 15) * 4;    // pixel quad 0,4,...,60
        const float* src = in + (size_t)c * npix + (size_t)(pix0 + xq);
#if USE_ASYNC_B128
        __builtin_amdgcn_global_load_async_to_lds_b128(
            (async_gv4i*)src, (async_lv4i*)&tile[c][xq], 0, 0);
        __builtin_amdgcn_s_wait_asynccnt(0);   // this wave's LDS writes done
#elif USE_ASYNC_LDS
        #pragma unroll
        for (int j = 0; j < 4; ++j) {
            __builtin_amdgcn_global_load_async_to_lds_b32(
                (async_gint*)(src + j), (async_lint*)&tile[c][xq + j], 0, 0);
        }
        __builtin_amdgcn_s_wait_asynccnt(0);
#else
        float4 v = *(const float4*)src;
        *(float4*)&tile[c][xq] = v;            // ds_store_b128
#endif
    }
    __syncthreads();

    // ---- outbound: transposed 4B reads from LDS, 16B coalesced store ----
    {
        const int x  = t >> 2;          // pixel 0..63
        const int c0 = (t & 3) * 4;     // channel quad 0,4,8,12
        float4 r;
        r.x = tile[c0 + 0][x];
        r.y = tile[c0 + 1][x];
        r.z = tile[c0 + 2][x];
        r.w = tile[c0 + 3][x];
        // out offset = (pix0 + x)*16 + c0 == pix0*16 + t*4  (contiguous 4KB)
        *(float4*)(outp + (size_t)pix0 * CHANNELS + (size_t)t * 4) = r;
    }
}

// ---------------------------------------------------------------------------
// Pass 2: gather. One lane = one sample. 8 taps, each tap = 4 x float4 loads
// from the HWC workspace (one 64B cache line per tap).
// ---------------------------------------------------------------------------
__device__ __forceinline__ void fma4(float4& a, float w, const float4 v) {
    a.x = fmaf(w, v.x, a.x);
    a.y = fmaf(w, v.y, a.y);
    a.z = fmaf(w, v.z, a.z);
    a.w = fmaf(w, v.w, a.w);
}

__global__ __launch_bounds__(256) void mip_gather_kernel(
        const float* __restrict__ uv, const float* __restrict__ p,
        const float* __restrict__ ws, float* __restrict__ out, int n) {
    int s = blockIdx.x * blockDim.x + threadIdx.x;
    if (s >= n) return;

    const float2 uvs = ((const float2*)uv)[s];
    const float u = uvs.x, v = uvs.y;

    float lf  = p[s] * (float)(NUM_LEVELS - 1);
    int   l0  = (int)floorf(lf);
    l0 = min(max(l0, 0), NUM_LEVELS - 1);
    int   l1  = min(l0 + 1, NUM_LEVELS - 1);
    float alpha = lf - (float)l0;

    float4 acc0 = make_float4(0.f, 0.f, 0.f, 0.f);
    float4 acc1 = acc0, acc2 = acc0, acc3 = acc0;

    const int   lvls[2] = {l0, l1};
    const float lwts[2] = {1.0f - alpha, alpha};

    #pragma unroll
    for (int k = 0; k < 2; ++k) {
        const int   lvl = lvls[k];
        const float lw  = lwts[k];
        const int   W   = 1024 >> lvl;
        const float wm1 = (float)(W - 1);

        float fx = fminf(fmaxf(u * wm1, 0.0f), wm1);
        float fy = fminf(fmaxf(v * wm1, 0.0f), wm1);
        float x0f = floorf(fx), y0f = floorf(fy);
        float wx = fx - x0f,    wy = fy - y0f;
        int x0 = (int)x0f,      y0 = (int)y0f;
        int x1 = min(x0 + 1, W - 1);
        int y1 = min(y0 + 1, W - 1);

        const float* __restrict__ base = ws + level_base(lvl);
        const float4* t00 = (const float4*)(base + ((size_t)y0 * W + x0) * CHANNELS);
        const float4* t01 = (const float4*)(base + ((size_t)y0 * W + x1) * CHANNELS);
        const float4* t10 = (const float4*)(base + ((size_t)y1 * W + x0) * CHANNELS);
        const float4* t11 = (const float4*)(base + ((size_t)y1 * W + x1) * CHANNELS);

        const float w00 = lw * (1.0f - wx) * (1.0f - wy);
        const float w01 = lw * wx * (1.0f - wy);
        const float w10 = lw * (1.0f - wx) * wy;
        const float w11 = lw * wx * wy;

        float4 a0 = t00[0], a1 = t00[1], a2 = t00[2], a3 = t00[3];
        float4 b0 = t01[0], b1 = t01[1], b2 = t01[2], b3 = t01[3];
        float4 c0 = t10[0], c1 = t10[1], c2 = t10[2], c3 = t10[3];
        float4 d0 = t11[0], d1 = t11[1], d2 = t11[2], d3 = t11[3];

        fma4(acc0, w00, a0); fma4(acc1, w00, a1); fma4(acc2, w00, a2); fma4(acc3, w00, a3);
        fma4(acc0, w01, b0); fma4(acc1, w01, b1); fma4(acc2, w01, b2); fma4(acc3, w01, b3);
        fma4(acc0, w10, c0); fma4(acc1, w10, c1); fma4(acc2, w10, c2); fma4(acc3, w10, c3);
        fma4(acc0, w11, d0); fma4(acc1, w11, d1); fma4(acc2, w11, d2); fma4(acc3, w11, d3);
    }

    float4* o = (float4*)(out + (size_t)s * CHANNELS);
    o[0] = acc0; o[1] = acc1; o[2] = acc2; o[3] = acc3;
}

// ---------------------------------------------------------------------------
// Fallback: direct CHW gather (only used if ws is too small). Correct but with
// 16x read amplification.
// ---------------------------------------------------------------------------
__global__ __launch_bounds__(256) void mip_gather_direct_kernel(
        const float* __restrict__ uv, const float* __restrict__ p,
        MipPtrs mips, float* __restrict__ out, int n) {
    int s = blockIdx.x * blockDim.x + threadIdx.x;
    if (s >= n) return;

    const float2 uvs = ((const float2*)uv)[s];
    const float u = uvs.x, v = uvs.y;

    float lf  = p[s] * (float)(NUM_LEVELS - 1);
    int   l0  = (int)floorf(lf);
    l0 = min(max(l0, 0), NUM_LEVELS - 1);
    int   l1  = min(l0 + 1, NUM_LEVELS - 1);
    float alpha = lf - (float)l0;

    float acc[CHANNELS];
    #pragma unroll
    for (int c = 0; c < CHANNELS; ++c) acc[c] = 0.0f;

    const int   lvls[2] = {l0, l1};
    const float lwts[2] = {1.0f - alpha, alpha};

    for (int k = 0; k < 2; ++k) {
        const int   lvl = lvls[k];
        const float lw  = lwts[k];
        const int   W   = 1024 >> lvl;
        const int   npix = W * W;
        const float wm1 = (float)(W - 1);

        float fx = fminf(fmaxf(u * wm1, 0.0f), wm1);
        float fy = fminf(fmaxf(v * wm1, 0.0f), wm1);
        float x0f = floorf(fx), y0f = floorf(fy);
        float wx = fx - x0f,    wy = fy - y0f;
        int x0 = (int)x0f,      y0 = (int)y0f;
        int x1 = min(x0 + 1, W - 1);
        int y1 = min(y0 + 1, W - 1);

        const float* tex = mips.m[lvl];
        const size_t o00 = (size_t)y0 * W + x0;
        const size_t o01 = (size_t)y0 * W + x1;
        const size_t o10 = (size_t)y1 * W + x0;
        const size_t o11 = (size_t)y1 * W + x1;

        const float w00 = lw * (1.0f - wx) * (1.0f - wy);
        const float w01 = lw * wx * (1.0f - wy);
        const float w10 = lw * (1.0f - wx) * wy;
        const float w11 = lw * wx * wy;

        #pragma unroll
        for (int c = 0; c < CHANNELS; ++c) {
            const size_t co = (size_t)c * npix;
            float r = w00 * tex[co + o00];
            r = fmaf(w01, tex[co + o01], r);
            r = fmaf(w10, tex[co + o10], r);
            r = fmaf(w11, tex[co + o11], r);
            acc[c] += r;
        }
    }
    #pragma unroll
    for (int c = 0; c < CHANNELS; ++c) out[(size_t)s * CHANNELS + c] = acc[c];
}

extern "C" void kernel_launch(void* const* d_in, const int* in_sizes, int n_in,
                              void* d_out, int out_size, void* d_ws, size_t ws_size,
                              hipStream_t stream) {
    const float* uv = (const float*)d_in[0];
    const float* p  = (const float*)d_in[1];
    MipPtrs mips;
    for (int i = 0; i < NUM_LEVELS; ++i) mips.m[i] = (const float*)d_in[2 + i];
    float* out = (float*)d_out;
    const int n = in_sizes[1];           // element count of p == N

    const size_t needed = WS_TOTAL_FLOATS * sizeof(float);
    if (ws_size >= needed) {
        float* ws = (float*)d_ws;
        mip_transpose_kernel<<<TOTAL_TILES, 256, 0, stream>>>(mips, ws);
        mip_gather_kernel<<<(n + 255) / 256, 256, 0, stream>>>(uv, p, ws, out, n);
    } else {
        mip_gather_direct_kernel<<<(n + 255) / 256, 256, 0, stream>>>(uv, p, mips, out, n);
    }
}